// RBFLayer_73349451481363
// MI455X (gfx1250) — compile-verified
//
#include <hip/hip_runtime.h>
#include <hip/hip_bf16.h>

// RBF layer: out = exp(-max(|x|^2 - 2 x.cT + |c|^2, 0) * shapes^2) @ W^T + b
// B=8192, K=256, D=128, O=8.  All f32.  Main GEMM via V_WMMA_F32_16X16X4_F32.
// v2: A-fragments hoisted to registers (LDS traffic halved), |x|^2 computed from
//     the A registers (x read from HBM exactly once), centers chunk double-buffered.

typedef float v2f __attribute__((ext_vector_type(2)));
typedef float v4f __attribute__((ext_vector_type(4)));
typedef float v8f __attribute__((ext_vector_type(8)));

#define RBF_B 8192
#define RBF_K 256
#define RBF_D 128
#define RBF_O 8

#define BM 64            // rows of x per block
#define THREADS 128      // 4 waves of 32
#define XS 132           // padded LDS stride (floats) for x tile
#define CS 132           // padded LDS stride (floats) for centers chunk

__global__ __launch_bounds__(THREADS)
void rbf_wmma_kernel(const float* __restrict__ x,
                     const float* __restrict__ centers,
                     const float* __restrict__ shapes,
                     const float* __restrict__ W,
                     const float* __restrict__ bias,
                     float* __restrict__ out)
{
    __shared__ float xlds[BM * XS];       // 33792 B  x tile (64 x 128, padded)
    __shared__ float clds[2][16 * CS];    // 16896 B  centers chunk, double buffered
    __shared__ float wlds[RBF_O * RBF_K]; //  8192 B  W, [o*256 + k]
    __shared__ float s2lds[RBF_K];        //  1024 B  shapes^2
    __shared__ float c2lds[RBF_K];        //  1024 B  |center|^2
    __shared__ float x2lds[BM];           //   256 B  |x row|^2

    const int tid  = threadIdx.x;
    const int lane = tid & 31;
    const int wave = tid >> 5;
    const int row0 = blockIdx.x * BM;

    // chunk-staging mapping (16 rows x 128 floats = 4 float4 per thread)
    const int crow = tid >> 3;            // 0..15
    const int cseg = tid & 7;             // 0..7 -> 16-float segment

    // ---------------- Phase 0: stage tiles ----------------
    // x tile: 64 rows x 32 float4 = 2048 float4, 16 per thread (coalesced)
    #pragma unroll
    for (int j = 0; j < 16; ++j) {
        int idx = tid + THREADS * j;
        int r = idx >> 5;
        int s = idx & 31;
        v4f v = *(const v4f*)(x + (size_t)(row0 + r) * RBF_D + s * 4);
        *(v4f*)(&xlds[r * XS + s * 4]) = v;
    }
    // W: 512 float4, 4 per thread
    #pragma unroll
    for (int j = 0; j < 4; ++j) {
        int idx = tid + THREADS * j;
        *(v4f*)(&wlds[idx * 4]) = *(const v4f*)(W + idx * 4);
    }
    // shapes^2
    #pragma unroll
    for (int j = 0; j < 2; ++j) {
        int i = tid + THREADS * j;
        float s = shapes[i];
        s2lds[i] = s * s;
    }
    // |center k|^2 for all 256 centers (2 rows per thread, from global/L2)
    #pragma unroll
    for (int j = 0; j < 2; ++j) {
        int r = tid + THREADS * j;
        float a2 = 0.0f;
        #pragma unroll 4
        for (int s = 0; s < 32; ++s) {
            v4f v = *(const v4f*)(centers + (size_t)r * RBF_D + s * 4);
            a2 += v.x * v.x + v.y * v.y + v.z * v.z + v.w * v.w;
        }
        c2lds[r] = a2;
    }
    // centers chunk 0 -> clds[0]
    {
        const float* src = centers + (size_t)crow * RBF_D + cseg * 16;
        float* dst = &clds[0][crow * CS + cseg * 16];
        #pragma unroll
        for (int u = 0; u < 4; ++u)
            *(v4f*)(dst + 4 * u) = *(const v4f*)(src + 4 * u);
    }

    __syncthreads();

    // ---------------- A fragments -> registers (invariant over K) ----------------
    // 32-bit A 16x4 layout: lane l -> row (l&15), k-pair 2*(l>>4) in vgpr {0,1}
    // 32-bit B 4x16 layout: lane l -> col (l&15), k-pair 2*(l>>4) in vgpr {0,1}
    const int n    = lane & 15;
    const int h    = lane >> 4;
    const int aOff = (wave * 16 + n) * XS + (h << 1);
    const int bOff = n * CS + (h << 1);

    v2f aFrag[32];
    #pragma unroll
    for (int d = 0; d < 32; ++d)
        aFrag[d] = *(const v2f*)(&xlds[aOff + 4 * d]);

    // |x row|^2 from the A registers: lanes n and n+16 hold complementary halves
    {
        float p = 0.0f;
        #pragma unroll
        for (int d = 0; d < 32; ++d)
            p += aFrag[d].x * aFrag[d].x + aFrag[d].y * aFrag[d].y;
        p += __shfl_xor(p, 16, 32);
        if (h == 0) x2lds[wave * 16 + n] = p;
    }

    __syncthreads();

    float acc[8][8];
    #pragma unroll
    for (int v = 0; v < 8; ++v)
        #pragma unroll
        for (int o = 0; o < 8; ++o) acc[v][o] = 0.0f;

    float x2r[8];
    #pragma unroll
    for (int v = 0; v < 8; ++v)
        x2r[v] = x2lds[wave * 16 + 8 * h + v];

    // ---------------- Main loop over 16 K-chunks (double buffered) ----------------
    for (int kc = 0; kc < 16; ++kc) {
        const int cur = kc & 1;
        const int nxt = cur ^ 1;

        // issue global loads for next chunk (land during compute phase)
        v4f pre[4];
        if (kc < 15) {
            const float* src = centers + (size_t)((kc + 1) * 16 + crow) * RBF_D + cseg * 16;
            #pragma unroll
            for (int u = 0; u < 4; ++u)
                pre[u] = *(const v4f*)(src + 4 * u);
        }

        // 16x16 xc tile: 32 WMMA f32 16x16x4, B fragments from LDS
        v8f c8 = {0.f, 0.f, 0.f, 0.f, 0.f, 0.f, 0.f, 0.f};
        #pragma unroll
        for (int d = 0; d < 32; ++d) {
            v2f bf = *(const v2f*)(&clds[cur][bOff + 4 * d]);
            c8 = __builtin_amdgcn_wmma_f32_16x16x4_f32(
                    false, aFrag[d], false, bf, (short)0, c8, false, false);
        }

        // epilogue: C layout -> lane holds rows m = 8*h + v (v = vgpr idx), col n
        const int   kg  = kc * 16 + n;
        const float s2  = s2lds[kg];
        const float cc2 = c2lds[kg];
        float wk[8];
        #pragma unroll
        for (int o = 0; o < 8; ++o) wk[o] = wlds[o * RBF_K + kg];

        #pragma unroll
        for (int v = 0; v < 8; ++v) {
            float sq = x2r[v] - 2.0f * c8[v] + cc2;
            sq = fmaxf(sq, 0.0f);
            float ph = __expf(-sq * s2);
            #pragma unroll
            for (int o = 0; o < 8; ++o)
                acc[v][o] = fmaf(ph, wk[o], acc[v][o]);
        }

        // store prefetched chunk into the alternate buffer
        if (kc < 15) {
            float* dst = &clds[nxt][crow * CS + cseg * 16];
            #pragma unroll
            for (int u = 0; u < 4; ++u)
                *(v4f*)(dst + 4 * u) = pre[u];
        }
        __syncthreads();
    }

    // ---------------- Cross-lane K reduction + store ----------------
    float bv[8];
    #pragma unroll
    for (int o = 0; o < 8; ++o) bv[o] = bias[o];

    #pragma unroll
    for (int v = 0; v < 8; ++v) {
        float r[8];
        #pragma unroll
        for (int o = 0; o < 8; ++o) {
            float t = acc[v][o];
            t += __shfl_xor(t, 1, 32);
            t += __shfl_xor(t, 2, 32);
            t += __shfl_xor(t, 4, 32);
            t += __shfl_xor(t, 8, 32);   // full sum within each 16-lane half
            r[o] = t;
        }
        if (n == 0) {                     // lanes 0 (rows 0..7) and 16 (rows 8..15)
            int rowg = row0 + wave * 16 + 8 * h + v;
            v4f lo = { r[0] + bv[0], r[1] + bv[1], r[2] + bv[2], r[3] + bv[3] };
            v4f hi = { r[4] + bv[4], r[5] + bv[5], r[6] + bv[6], r[7] + bv[7] };
            *(v4f*)(out + (size_t)rowg * RBF_O)     = lo;
            *(v4f*)(out + (size_t)rowg * RBF_O + 4) = hi;
        }
    }
}

extern "C" void kernel_launch(void* const* d_in, const int* in_sizes, int n_in,
                              void* d_out, int out_size, void* d_ws, size_t ws_size,
                              hipStream_t stream) {
    const float* x       = (const float*)d_in[0];   // (8192,128)
    const float* centers = (const float*)d_in[1];   // (256,128)
    const float* shapes  = (const float*)d_in[2];   // (256,)
    const float* W       = (const float*)d_in[3];   // (8,256)
    const float* b       = (const float*)d_in[4];   // (8,)
    float* out = (float*)d_out;                     // (8192,8)

    dim3 grid(RBF_B / BM);   // 128 blocks
    dim3 block(THREADS);     // 4 waves
    rbf_wmma_kernel<<<grid, block, 0, stream>>>(x, centers, shapes, W, b, out);
}